// GlobalAttention_18957985645094
// MI455X (gfx1250) — compile-verified
//
#include <hip/hip_runtime.h>
#include <math.h>
#include <stdint.h>

// ---------------------------------------------------------------------------
// GlobalAttention (MLP/additive attention), B=16, Lq=Lc=256, D=64, fp32.
// Heavy part is 67M tanh (VALU TRANS); all GEMMs use V_WMMA_F32_16X16X4_F32.
// sc chunks staged with GLOBAL_LOAD_ASYNC_TO_LDS_B128 + ASYNCcnt double-buffer.
// d_in order: query, bank, mask, Wq, Wc, bc, Ws, bs, Woq, Woc, boc
// d_out: output[16*256*64] floats ++ attn[16*256*256] floats
// ---------------------------------------------------------------------------

typedef __attribute__((ext_vector_type(2))) float v2f;
typedef __attribute__((ext_vector_type(8))) float v8f;

#define BATCH 16
#define LQ 256
#define LC 256
#define DIM 64
#define OUT_OFFSET (BATCH * LQ * DIM)   // where attn starts in d_out

#define SQ_STRIDE 68     // floats; bank stride 4 -> 16 rows hit 16 banks; 16B aligned
#define SCORE_STRIDE 258 // floats; even -> 8B-aligned WMMA pair loads, conflict-free

// ---- fast tanh: hardware v_tanh_f32 if available, else exp2+rcp ----------
__device__ __forceinline__ float fast_tanh(float x) {
#if __has_builtin(__builtin_amdgcn_tanhf)
  return __builtin_amdgcn_tanhf(x);
#elif __has_builtin(__builtin_amdgcn_tanh_f32)
  return __builtin_amdgcn_tanh_f32(x);
#else
  float xc = fminf(10.f, fmaxf(-10.f, x));
  float e2 = __builtin_amdgcn_exp2f(xc * 2.8853900817779268f); // e^(2x)
  return (e2 - 1.f) * __builtin_amdgcn_rcpf(e2 + 1.f);
#endif
}

// ---- CDNA5 async global->LDS copy (ASYNCcnt), 16 bytes per lane -----------
__device__ __forceinline__ void async_copy_b128(uint32_t lds_off,
                                                const float* gptr) {
  asm volatile("global_load_async_to_lds_b128 %0, %1, off"
               :: "v"(lds_off), "v"(gptr)
               : "memory");
}
__device__ __forceinline__ void wait_asynccnt_0() {
  asm volatile("s_wait_asynccnt 0x0" ::: "memory");
}
__device__ __forceinline__ void wait_asynccnt_1() {
  asm volatile("s_wait_asynccnt 0x1" ::: "memory");
}

// ---- 16x16 fp32 output tile of X[16 x 64] * W^T (W is [64x64] row-major,
//      W[e][d]); X row-major stride 64. K=64 in 16 WMMA_F32_16X16X4 steps.
//      A layout: lane L -> M=L%16, VGPR j holds K = kb + 2*(L/16) + j
//      B layout (mirror): lane L -> N=L%16, VGPR j holds K = kb + 2*(L/16) + j
__device__ __forceinline__ v8f gemm16_xwT(v8f acc,
                                          const float* __restrict__ X,
                                          const float* __restrict__ W,
                                          int n0, int lane) {
  const int mn = lane & 15;
  const int kh = (lane >> 4) << 1;
#pragma unroll
  for (int kb = 0; kb < 64; kb += 4) {
    const int k = kb + kh;
    v2f a, bv;
    a.x = X[mn * 64 + k];
    a.y = X[mn * 64 + k + 1];
    bv.x = W[(n0 + mn) * 64 + k];
    bv.y = W[(n0 + mn) * 64 + k + 1];
    acc = __builtin_amdgcn_wmma_f32_16x16x4_f32(false, a, false, bv,
                                                (short)0, acc, false, false);
  }
  return acc;
}

// ===========================================================================
// Kernel 1: sq = query*Wq^T ; sc = bank*Wc^T + bc.
// 8192 row-tiles-of-16 total rows => 512 blocks; block=128 (4 waves = N 0..63)
// ===========================================================================
__global__ void ga_proj_kernel(const float* __restrict__ query,
                               const float* __restrict__ bank,
                               const float* __restrict__ Wq,
                               const float* __restrict__ Wc,
                               const float* __restrict__ bc,
                               float* __restrict__ sq,
                               float* __restrict__ sc) {
  const int tile = blockIdx.x;          // 0..511
  const bool is_c = tile >= 256;
  const int row0 = (tile & 255) * 16;   // row tile within 4096 rows
  const int wave = threadIdx.x >> 5;
  const int lane = threadIdx.x & 31;
  const int n0 = wave * 16;

  const float* X = (is_c ? bank : query) + row0 * DIM;
  const float* W = is_c ? Wc : Wq;
  float* Y = is_c ? sc : sq;

  v8f c = {};
  c = gemm16_xwT(c, X, W, n0, lane);

  const int n = n0 + (lane & 15);
  const float bias = is_c ? bc[n] : 0.f;
#pragma unroll
  for (int r = 0; r < 8; ++r) {
    const int m = r + ((lane >> 4) << 3);
    Y[(row0 + m) * DIM + n] = c[r] + bias;
  }
}

// ===========================================================================
// Kernel 2: scores + softmax + attn output + context = attn * bank (WMMA).
// grid = B * (Lq/16) = 256 blocks ; block = 256 threads (8 waves).
// sc chunks double-buffered via async global->LDS copies (ASYNCcnt).
// ===========================================================================
__global__ void ga_attn_kernel(const float* __restrict__ sq,
                               const float* __restrict__ sc,
                               const float* __restrict__ bank,
                               const unsigned char* __restrict__ mask,
                               const float* __restrict__ Ws,
                               const float* __restrict__ bs,
                               float* __restrict__ attn_out, // d_out + OUT_OFFSET
                               float* __restrict__ ctx) {    // [B*Lq, 64] ws
  __shared__ __align__(16) float sq_s[16][SQ_STRIDE];
  __shared__ __align__(16) float sc_db[2][16][SQ_STRIDE];  // async double buffer
  __shared__ __align__(16) float ws_s[64];
  __shared__ float score_s[16][SCORE_STRIDE];
  __shared__ float red_s[16][17];

  const int b  = blockIdx.x >> 4;
  const int q0 = (blockIdx.x & 15) * 16;
  const int t  = threadIdx.x;

  // per-thread staging coordinates: one float4 (b128) per thread per tile
  const int sr = t >> 4;          // row 0..15
  const int sd = (t & 15) * 4;    // col 0..60 step 4

  // kick off async copy of sc chunk 0 into buffer 0
  {
    const float* g0 = sc + (b * LC + 0 * 16 + sr) * DIM + sd;
    async_copy_b128((uint32_t)(uintptr_t)&sc_db[0][sr][sd], g0);
  }

  // stage the 16-row sq tile (vectorized) and Ws while chunk 0 is in flight
  *(float4*)&sq_s[sr][sd] =
      *(const float4*)&sq[(b * LQ + q0 + sr) * DIM + sd];
  if (t < 64) ws_s[t] = Ws[t];
  const float bsv = bs[0];

  // ---- score phase: thread owns (qi, cj); 16 c-chunks of 16 ----
  const int qi = t >> 4;
  const int cj = t & 15;
  const float4* sq4 = (const float4*)&sq_s[qi][0];
  const float4* ws4 = (const float4*)&ws_s[0];

  for (int ct = 0; ct < 16; ++ct) {
    if (ct < 15) {  // prefetch next chunk into the other buffer
      const float* gn = sc + (b * LC + (ct + 1) * 16 + sr) * DIM + sd;
      async_copy_b128((uint32_t)(uintptr_t)&sc_db[(ct + 1) & 1][sr][sd], gn);
      wait_asynccnt_1();   // chunk ct complete (async loads retire in order)
    } else {
      wait_asynccnt_0();
    }
    __syncthreads();       // chunk ct visible to whole block

    const float4* sc4 = (const float4*)&sc_db[ct & 1][cj][0];
    float s0 = 0.f, s1 = 0.f, s2 = 0.f, s3 = 0.f;
#pragma unroll
    for (int d4 = 0; d4 < 16; ++d4) {
      const float4 a = sq4[d4];
      const float4 c4 = sc4[d4];
      const float4 w = ws4[d4];
      s0 += w.x * fast_tanh(a.x + c4.x);
      s1 += w.y * fast_tanh(a.y + c4.y);
      s2 += w.z * fast_tanh(a.z + c4.z);
      s3 += w.w * fast_tanh(a.w + c4.w);
    }
    float s = bsv + ((s0 + s1) + (s2 + s3));

    const int c = ct * 16 + cj;
    if (mask[(b * LQ + q0 + qi) * LC + c]) s = -INFINITY;
    score_s[qi][c] = s;
    __syncthreads();       // all reads of this buffer done before overwrite
  }

  // ---- softmax: 16 threads per row, strided over 256 columns ----
  {
    const int r = t >> 4;
    const int g = t & 15;
    float mx = -INFINITY;
#pragma unroll
    for (int j = 0; j < 16; ++j) mx = fmaxf(mx, score_s[r][g + j * 16]);
    red_s[r][g] = mx;
    __syncthreads();
    float rowmax = -INFINITY;
#pragma unroll
    for (int k = 0; k < 16; ++k) rowmax = fmaxf(rowmax, red_s[r][k]);

    float sum = 0.f;
#pragma unroll
    for (int j = 0; j < 16; ++j) {
      const int c = g + j * 16;
      const float e = __builtin_amdgcn_exp2f((score_s[r][c] - rowmax) * 1.44269504f);
      score_s[r][c] = e;
      sum += e;
    }
    __syncthreads();
    red_s[r][g] = sum;
    __syncthreads();
    float rowsum = 0.f;
#pragma unroll
    for (int k = 0; k < 16; ++k) rowsum += red_s[r][k];
    const float inv = __builtin_amdgcn_rcpf(rowsum);
#pragma unroll
    for (int j = 0; j < 16; ++j) {
      const int c = g + j * 16;
      const float p = score_s[r][c] * inv;
      score_s[r][c] = p;                                  // keep for WMMA
      attn_out[(b * LQ + q0 + r) * LC + c] = p;           // output #2
    }
  }
  __syncthreads();

  // ---- context = attn[16x256] * bank[b][256x64] with WMMA f32 16x16x4 ----
  const int wave = t >> 5;
  const int lane = t & 31;
  if (wave < 4) {                       // wave-uniform branch: EXEC all-1s
    const int n0 = wave * 16;
    const int mn = lane & 15;
    const int kh = (lane >> 4) << 1;
    v8f c = {};
#pragma unroll 4
    for (int kb = 0; kb < LC; kb += 4) {
      const int k = kb + kh;
      v2f a, bv;
      a.x = score_s[mn][k];
      a.y = score_s[mn][k + 1];
      bv.x = bank[(b * LC + k) * DIM + n0 + mn];
      bv.y = bank[(b * LC + k + 1) * DIM + n0 + mn];
      c = __builtin_amdgcn_wmma_f32_16x16x4_f32(false, a, false, bv,
                                                (short)0, c, false, false);
    }
#pragma unroll
    for (int r = 0; r < 8; ++r) {
      const int m = r + ((lane >> 4) << 3);
      ctx[(b * LQ + q0 + m) * DIM + n0 + mn] = c[r];
    }
  }
}

// ===========================================================================
// Kernel 3: output = ctx*Woc^T + boc + query*Woq^T  (two GEMMs, one acc).
// 4096/16 = 256 blocks; block = 128 (4 waves cover N=0..63).
// ===========================================================================
__global__ void ga_outproj_kernel(const float* __restrict__ ctx,
                                  const float* __restrict__ query,
                                  const float* __restrict__ Woc,
                                  const float* __restrict__ boc,
                                  const float* __restrict__ Woq,
                                  float* __restrict__ out) {
  const int row0 = blockIdx.x * 16;
  const int wave = threadIdx.x >> 5;
  const int lane = threadIdx.x & 31;
  const int n0 = wave * 16;

  v8f c = {};
  c = gemm16_xwT(c, ctx + row0 * DIM, Woc, n0, lane);
  c = gemm16_xwT(c, query + row0 * DIM, Woq, n0, lane);

  const int n = n0 + (lane & 15);
  const float bias = boc[n];
#pragma unroll
  for (int r = 0; r < 8; ++r) {
    const int m = r + ((lane >> 4) << 3);
    out[(row0 + m) * DIM + n] = c[r] + bias;
  }
}

// ===========================================================================
extern "C" void kernel_launch(void* const* d_in, const int* in_sizes, int n_in,
                              void* d_out, int out_size, void* d_ws, size_t ws_size,
                              hipStream_t stream) {
  const float* query = (const float*)d_in[0];
  const float* bank  = (const float*)d_in[1];
  const unsigned char* mask = (const unsigned char*)d_in[2];
  const float* Wq  = (const float*)d_in[3];
  const float* Wc  = (const float*)d_in[4];
  const float* bc  = (const float*)d_in[5];
  const float* Ws  = (const float*)d_in[6];
  const float* bs  = (const float*)d_in[7];
  const float* Woq = (const float*)d_in[8];
  const float* Woc = (const float*)d_in[9];
  const float* boc = (const float*)d_in[10];

  float* out      = (float*)d_out;
  float* attn_out = out + OUT_OFFSET;

  // workspace: sq | sc | ctx, each B*256*64 floats (1 MB) => 3 MB total
  float* sq  = (float*)d_ws;
  float* sc  = sq + BATCH * LQ * DIM;
  float* ctx = sc + BATCH * LC * DIM;

  ga_proj_kernel<<<512, 128, 0, stream>>>(query, bank, Wq, Wc, bc, sq, sc);
  ga_attn_kernel<<<BATCH * (LQ / 16), 256, 0, stream>>>(sq, sc, bank, mask,
                                                        Ws, bs, attn_out, ctx);
  ga_outproj_kernel<<<BATCH * LQ / 16, 128, 0, stream>>>(ctx, query, Woc, boc,
                                                         Woq, out);
}